// DSAFTNKSPLLoss_7799660609717
// MI455X (gfx1250) — compile-verified
//
#include <hip/hip_runtime.h>
#include <hip/hip_bf16.h>
#include <math.h>

#define N 8192
#define K2_BLOCK 256
#define K2_WAVES (K2_BLOCK / 32)          // 8 waves per block
#define NTILES (N / 16)                   // 512 row tiles
#define K2_GRID (NTILES / K2_WAVES)       // 64 blocks

typedef __attribute__((ext_vector_type(2))) float v2f;
typedef __attribute__((ext_vector_type(8))) float v8f;

// ---------------------------------------------------------------------------
// Kernel 0: residuals  e_a = log(dur_a + eps) - theta_a
// ---------------------------------------------------------------------------
__global__ __launch_bounds__(256)
void dsaft_prep_kernel(const float* __restrict__ log_h,
                       const float* __restrict__ dur,
                       float* __restrict__ eraw) {
  const int i = blockIdx.x * 256 + threadIdx.x;
  eraw[i] = logf(dur[i] + 1e-32f) - log_h[i];
}

// ---------------------------------------------------------------------------
// Kernel 1: O(n^2) rank = stable argsort position (lexicographic (key, idx)
// tie-break => bijective ranks, identical to jnp.argsort semantics).
// rank(a) == inv[a]; directly emits:
//   ws_e[rank] = e_a                (e_sorted, scatter)
//   ws_ev[a]   = events[rank(a)]    (= events[inv][a], the reference quirk)
//   ws_th[a]   = log_h[rank(a)]
// Fully parallel across the device, unlike a single-block bitonic sort.
// ---------------------------------------------------------------------------
__global__ __launch_bounds__(256)
void dsaft_rank_kernel(const float* __restrict__ eraw,
                       const float* __restrict__ log_h,
                       const int*   __restrict__ events,
                       float* __restrict__ ws_e,
                       float* __restrict__ ws_ev,
                       float* __restrict__ ws_th) {
  __shared__ float se[N];   // 32 KB
  for (int i = threadIdx.x; i < N; i += 256) se[i] = eraw[i];
  __syncthreads();

  const int a = blockIdx.x * 256 + threadIdx.x;   // one row per thread
  const float ei = se[a];
  int cnt = 0;
  const float4* se4 = (const float4*)se;
#pragma unroll 4
  for (int q = 0; q < N / 4; ++q) {
    const float4 v = se4[q];
    const int j = q * 4;
    cnt += (int)((v.x < ei) || (v.x == ei && (j + 0) < a));
    cnt += (int)((v.y < ei) || (v.y == ei && (j + 1) < a));
    cnt += (int)((v.z < ei) || (v.z == ei && (j + 2) < a));
    cnt += (int)((v.w < ei) || (v.w == ei && (j + 3) < a));
  }
  ws_e[cnt] = ei;                    // sorted residuals
  ws_ev[a]  = (float)events[cnt];    // events[inv[a]]
  ws_th[a]  = log_h[cnt];            // theta[inv[a]]
}

// ---------------------------------------------------------------------------
// Kernel 2: pairwise pass. One wave per 16-row i-tile; WMMA F32 16x16x4
// generates diff tiles  D[m][n] = e_row[m]*1 + 1*(-e_col[n]).
// Branch-free Phi via Abramowitz-Stegun 7.1.26 sharing the Gaussian exp:
//   g = exp(-0.5 d^2)  (= pdf numerator AND e^{-x^2} with x = |d|/sqrt2)
//   t = 1/(1 + p x);  erf(x) = 1 - poly5(t) * g
// => 1 v_exp + 1 v_rcp + ~10 VALU per pair, no EXEC divergence.
// ---------------------------------------------------------------------------
__global__ __launch_bounds__(K2_BLOCK)
void dsaft_pair_kernel(const float* __restrict__ ws_e,
                       const float* __restrict__ ws_ev,
                       const float* __restrict__ ws_th,
                       float* __restrict__ ws_part) {
  __shared__ float se[N];    // 32 KB
  __shared__ float sev[N];   // 32 KB
  for (int i = threadIdx.x; i < N; i += K2_BLOCK) {
    se[i]  = ws_e[i];
    sev[i] = ws_ev[i];
  }
  __syncthreads();

  const int lane = threadIdx.x & 31;
  const int wave = threadIdx.x >> 5;
  const int tile = blockIdx.x * K2_WAVES + wave;   // 0..511
  const int i0   = tile * 16;
  const bool lo  = lane < 16;

  // A (16x4 f32): lanes 0-15 hold K=0 (e_row), K=1 (1.0); lanes 16-31 zero.
  v2f a;
  a.x = lo ? se[i0 + lane] : 0.0f;
  a.y = lo ? 1.0f : 0.0f;

  float accp[8], accc[8];
#pragma unroll
  for (int r = 0; r < 8; ++r) { accp[r] = 0.0f; accc[r] = 0.0f; }

  const float INV_SQRT2 = 0.70710678118654752f;
  const float PP = 0.3275911f;
  const float A1 = 0.254829592f, A2 = -0.284496736f, A3 = 1.421413741f,
              A4 = -1.453152027f, A5 = 1.061405429f;

  const v8f c0 = {};
  for (int j0 = 0; j0 < N; j0 += 16) {
    const float ej  = se[j0 + (lane & 15)];
    const float evj = sev[j0 + (lane & 15)];
    // B (4x16 f32): row K=0 all 1, row K=1 = -e_col; K=2,3 zero.
    v2f b;
    b.x = lo ? 1.0f : 0.0f;
    b.y = lo ? -ej  : 0.0f;
    // D[m][n] = e_row[m] - e_col[n]
    v8f d = __builtin_amdgcn_wmma_f32_16x16x4_f32(
        false, a, false, b, (short)0, c0, false, false);
#pragma unroll
    for (int r = 0; r < 8; ++r) {
      const float t  = d[r];
      const float ax = fabsf(t) * INV_SQRT2;
      const float g  = __expf(-ax * ax);                 // exp(-0.5 t^2)
      const float rt = __builtin_amdgcn_rcpf(fmaf(PP, ax, 1.0f));
      const float y  = rt * (A1 + rt * (A2 + rt * (A3 + rt * (A4 + rt * A5)))) * g;
      const float h  = 0.5f * y;
      accp[r] += g * evj;                                // pdf * ev (unnormalized)
      accc[r] += (t >= 0.0f) ? (1.0f - h) : h;           // Phi(t)
    }
  }

  // Reduce along N within each 16-lane half (C/D layout:
  // lanes 0-15 -> rows i0+r, lanes 16-31 -> rows i0+8+r).
#pragma unroll
  for (int r = 0; r < 8; ++r) {
#pragma unroll
    for (int m = 1; m <= 8; m <<= 1) {
      accp[r] += __shfl_xor(accp[r], m, 32);
      accc[r] += __shfl_xor(accc[r], m, 32);
    }
  }

  if ((lane & 15) == 0) {
    const float invn     = 1.0f / (float)N;
    const float pdfscale = 0.3989422804014327f * invn;  // 1/(sqrt(2pi)*n*an)
    const float epsterm  = (float)N * 1e-32f;           // n * 1e-32 per-elem eps
    const int rbase = i0 + (lane >> 4) * 8;
    float s = 0.0f;
#pragma unroll
    for (int r = 0; r < 8; ++r) {
      const int row = rbase + r;
      const float cond = accp[r] * pdfscale + epsterm;
      const float sv   = accc[r] * invn;
      s += (logf(cond) - logf(sv) + ws_th[row]) * ws_ev[row];
    }
    ws_part[tile * 2 + (lane >> 4)] = s;   // 1024 partials total
  }
}

// ---------------------------------------------------------------------------
// Kernel 3: deterministic fixed-tree reduction of 1024 partials.
// ---------------------------------------------------------------------------
__global__ __launch_bounds__(256)
void dsaft_reduce_kernel(const float* __restrict__ ws_part,
                         float* __restrict__ out) {
  __shared__ float buf[256];
  const int t = threadIdx.x;
  float s = 0.0f;
  for (int i = t; i < 2 * NTILES; i += 256) s += ws_part[i];
  buf[t] = s;
  __syncthreads();
  for (int w = 128; w > 0; w >>= 1) {
    if (t < w) buf[t] += buf[t + w];
    __syncthreads();
  }
  if (t == 0) out[0] = -buf[0] / (float)N;
}

// ---------------------------------------------------------------------------
extern "C" void kernel_launch(void* const* d_in, const int* in_sizes, int n_in,
                              void* d_out, int out_size, void* d_ws, size_t ws_size,
                              hipStream_t stream) {
  const float* log_h  = (const float*)d_in[0];   // (8192,1) fp32
  const float* dur    = (const float*)d_in[1];   // (8192,)  fp32
  const int*   events = (const int*)d_in[2];     // (8192,)  int32
  float* out = (float*)d_out;                    // scalar fp32 loss

  float* ws      = (float*)d_ws;
  float* ws_e    = ws;                 // 8192 f32  sorted residuals
  float* ws_ev   = ws + N;             // 8192 f32  events[inv]
  float* ws_th   = ws + 2 * N;         // 8192 f32  theta[inv]
  float* ws_eraw = ws + 3 * N;         // 8192 f32  unsorted residuals
  float* ws_part = ws + 4 * N;         // 1024 f32  per-wave partials

  dsaft_prep_kernel<<<N / 256, 256, 0, stream>>>(log_h, dur, ws_eraw);

  dsaft_rank_kernel<<<N / 256, 256, 0, stream>>>(
      ws_eraw, log_h, events, ws_e, ws_ev, ws_th);

  dsaft_pair_kernel<<<K2_GRID, K2_BLOCK, 0, stream>>>(
      ws_e, ws_ev, ws_th, ws_part);

  dsaft_reduce_kernel<<<1, 256, 0, stream>>>(ws_part, out);
}